// ST_EIConvSPGRAD2_55662776156166
// MI455X (gfx1250) — compile-verified
//
#include <hip/hip_runtime.h>
#include <hip/hip_bf16.h>

typedef float v2f __attribute__((ext_vector_type(2)));
typedef float v8f __attribute__((ext_vector_type(8)));

__device__ __forceinline__ v8f wmma_f32_16x16x4(v2f a, v2f b, v8f c) {
  // 8 args: (neg_a, A, neg_b, B, c_mod, C, reuse_a, reuse_b)
  return __builtin_amdgcn_wmma_f32_16x16x4_f32(false, a, false, b, (short)0, c,
                                               false, false);
}

// ---------------- zero accumulators ----------------
__global__ void zero_kernel(float* __restrict__ p, int count) {
  int i = blockIdx.x * blockDim.x + threadIdx.x;
  if (i < count) p[i] = 0.0f;
}

// ---------------- emh = relu(concat(h_feat, e_feat) @ W_trans[48,32]) ----------------
// 256 threads = 8 waves per block; W_trans staged once per block in LDS,
// pair-interleaved so each B fragment is a single aligned ds_load_b64.
__global__ void __launch_bounds__(256) emh_kernel(
    const float* __restrict__ h_feat,
    const float* __restrict__ e_feat,
    const float* __restrict__ W,  // [48,32] row-major
    float* __restrict__ emh, int n) {
  __shared__ float Wl[24 * 64];  // [(k>>1)*64 + col*2 + (k&1)], 6144 B
  for (int idx = threadIdx.x; idx < 48 * 32; idx += 256) {
    int k = idx >> 5;
    int nn = idx & 31;
    Wl[(k >> 1) * 64 + nn * 2 + (k & 1)] = W[idx];
  }
  __syncthreads();  // all threads reach this before any wave exits

  int wave = (blockIdx.x * blockDim.x + threadIdx.x) >> 5;
  int lane = threadIdx.x & 31;
  int ntiles = (n + 15) >> 4;
  if (wave >= ntiles) return;  // wave-uniform; no barriers after this point

  int row0 = wave << 4;
  int row = row0 + (lane & 15);
  if (row >= n) row = n - 1;     // clamp loads; EXEC stays all-ones for WMMA
  int khalf = (lane >> 4) << 1;  // lanes 0-15 -> K {0,1}; lanes 16-31 -> K {2,3}
  int col = lane & 15;

  v8f c0 = {0.f, 0.f, 0.f, 0.f, 0.f, 0.f, 0.f, 0.f};
  v8f c1 = {0.f, 0.f, 0.f, 0.f, 0.f, 0.f, 0.f, 0.f};

#pragma unroll
  for (int k0 = 0; k0 < 48; k0 += 4) {
    int ka = k0 + khalf;  // always even
    v2f a;
    // concat(h_feat[32], e_feat[16]); segment choice folds under full unroll
    a.x = (ka < 32) ? h_feat[row * 32 + ka] : e_feat[row * 16 + (ka - 32)];
    a.y = (ka + 1 < 32) ? h_feat[row * 32 + ka + 1] : e_feat[row * 16 + (ka + 1 - 32)];
    int pb = (ka >> 1) * 64;
    v2f b0 = *(const v2f*)&Wl[pb + col * 2];
    v2f b1 = *(const v2f*)&Wl[pb + (col + 16) * 2];
    c0 = wmma_f32_16x16x4(a, b0, c0);
    c1 = wmma_f32_16x16x4(a, b1, c1);
  }

  int rsel = (lane >> 4) << 3;  // lanes 16-31 hold rows r+8
#pragma unroll
  for (int r = 0; r < 8; ++r) {
    int rr = row0 + rsel + r;
    if (rr < n) {
      emh[rr * 32 + col]      = fmaxf(c0[r], 0.f);
      emh[rr * 32 + 16 + col] = fmaxf(c1[r], 0.f);
    }
  }
}

// ---------------- edge pass 1: h_sum scatter + first-order grads ----------------
// One wave per edge (xx edges first, then yy). 32 lanes: emh row scatter-add;
// lanes 0-15 additionally: w * (e[src] - e[dst]) scatter-add.
__global__ void edge_pass1(const float* __restrict__ emh,
                           const float* __restrict__ e_feat,
                           const int* __restrict__ src_xx, const int* __restrict__ dst_xx,
                           const float* __restrict__ w_xx,
                           const int* __restrict__ src_yy, const int* __restrict__ dst_yy,
                           const float* __restrict__ w_yy,
                           float* __restrict__ h_sum,
                           float* __restrict__ gx, float* __restrict__ gy, int E) {
  long long tid = (long long)blockIdx.x * blockDim.x + threadIdx.x;
  long long e = tid >> 5;
  int lane = threadIdx.x & 31;
  if (e >= 2LL * E) return;

  const int* S; const int* D; const float* W; float* G; int ee;
  if (e < E) { S = src_xx; D = dst_xx; W = w_xx; G = gx; ee = (int)e; }
  else       { S = src_yy; D = dst_yy; W = w_yy; G = gy; ee = (int)(e - E); }

  int s = S[ee];
  int d = D[ee];
  atomicAdd(&h_sum[d * 32 + lane], emh[s * 32 + lane]);
  if (lane < 16) {
    float wv = W[ee];
    atomicAdd(&G[d * 16 + lane], wv * (e_feat[s * 16 + lane] - e_feat[d * 16 + lane]));
  }
}

// ---------------- edge pass 2: second-order grads (needs pass 1 complete) ----------------
__global__ void edge_pass2(const float* __restrict__ gx, const float* __restrict__ gy,
                           const int* __restrict__ src_xx, const int* __restrict__ dst_xx,
                           const float* __restrict__ w_xx,
                           const int* __restrict__ src_yy, const int* __restrict__ dst_yy,
                           const float* __restrict__ w_yy,
                           float* __restrict__ g2x, float* __restrict__ g2y, int E) {
  long long tid = (long long)blockIdx.x * blockDim.x + threadIdx.x;
  long long e = tid >> 4;
  int f = threadIdx.x & 15;
  if (e >= 2LL * E) return;

  const int* S; const int* D; const float* W; const float* G1; float* G2; int ee;
  if (e < E) { S = src_xx; D = dst_xx; W = w_xx; G1 = gx; G2 = g2x; ee = (int)e; }
  else       { S = src_yy; D = dst_yy; W = w_yy; G1 = gy; G2 = g2y; ee = (int)(e - E); }

  int s = S[ee];
  int d = D[ee];
  atomicAdd(&G2[d * 16 + f], W[ee] * (G1[s * 16 + f] - G1[d * 16 + f]));
}

// ---------------- h_cat element gather (branches fold under unroll) ----------------
__device__ __forceinline__ float hcat_load(const float* __restrict__ h_sum,
                                           const float* __restrict__ gx,
                                           const float* __restrict__ gy,
                                           const float* __restrict__ g2x,
                                           const float* __restrict__ g2y,
                                           int row, int k) {
  if (k < 32) return fmaxf(h_sum[row * 32 + k], 0.f);  // h = relu(h_sum)
  if (k < 48) return gx[row * 16 + (k - 32)];
  if (k < 64) return gy[row * 16 + (k - 48)];
  if (k < 80) return g2x[row * 16 + (k - 64)];
  return g2y[row * 16 + (k - 80)];
}

// ---------------- final: rain_h = relu(h_cat @ W_rin[96,128]); rainfall; h out ----------------
// 256 threads = 8 waves per block; W_rin staged once per block in LDS (48 KB),
// B fragments read as ds_load_b64; W_rout head reduced in registers via shfl.
__global__ void __launch_bounds__(256) rain_kernel(
    const float* __restrict__ h_sum,
    const float* __restrict__ gx, const float* __restrict__ gy,
    const float* __restrict__ g2x, const float* __restrict__ g2y,
    const float* __restrict__ W_rin,   // [96,128]
    const float* __restrict__ W_rout,  // [128]
    const float* __restrict__ rain0,
    float* __restrict__ out_rain,      // [n]
    float* __restrict__ out_h,         // [n,32]
    int n) {
  __shared__ float Wl[48 * 256];  // [(k>>1)*256 + col*2 + (k&1)], 49152 B
  for (int idx = threadIdx.x; idx < 96 * 128; idx += 256) {
    int k = idx >> 7;
    int nn = idx & 127;
    Wl[(k >> 1) * 256 + nn * 2 + (k & 1)] = W_rin[idx];
  }
  __syncthreads();  // only barrier; all threads reach it

  int wave = (blockIdx.x * blockDim.x + threadIdx.x) >> 5;
  int lane = threadIdx.x & 31;
  int ntiles = (n + 15) >> 4;
  if (wave >= ntiles) return;  // wave-uniform exit after the barrier

  int row0 = wave << 4;
  int row = row0 + (lane & 15);
  if (row >= n) row = n - 1;
  int khalf = (lane >> 4) << 1;
  int col = lane & 15;

  // h output: relu of accumulated neighbor sums (uniform predicates only)
  for (int r = 0; r < 16; ++r) {
    int rr = row0 + r;
    if (rr < n) out_h[rr * 32 + lane] = fmaxf(h_sum[rr * 32 + lane], 0.f);
  }

  // hoist W_rout gather: this lane's 8 head weights
  float wr[8];
#pragma unroll
  for (int t = 0; t < 8; ++t) wr[t] = W_rout[t * 16 + col];

  v8f c[8];
  v8f zero = {0.f, 0.f, 0.f, 0.f, 0.f, 0.f, 0.f, 0.f};
#pragma unroll
  for (int t = 0; t < 8; ++t) c[t] = zero;

#pragma unroll
  for (int k0 = 0; k0 < 96; k0 += 4) {
    int ka = k0 + khalf;  // always even
    v2f a;
    a.x = hcat_load(h_sum, gx, gy, g2x, g2y, row, ka);
    a.y = hcat_load(h_sum, gx, gy, g2x, g2y, row, ka + 1);
    int pb = (ka >> 1) * 256;
#pragma unroll
    for (int t = 0; t < 8; ++t) {
      v2f b = *(const v2f*)&Wl[pb + (t * 16 + col) * 2];
      c[t] = wmma_f32_16x16x4(a, b, c[t]);
    }
  }

  // head: rainfall[row] = rain0[row] + relu(rain_h[row,:]) . W_rout
  // D layout: c[t][r] = rain_h[row0 + r + 8*(lane>>4)][t*16 + col]
#pragma unroll
  for (int r = 0; r < 8; ++r) {
    float s = 0.f;
#pragma unroll
    for (int t = 0; t < 8; ++t) s += fmaxf(c[t][r], 0.f) * wr[t];
    s += __shfl_xor(s, 1, 32);
    s += __shfl_xor(s, 2, 32);
    s += __shfl_xor(s, 4, 32);
    s += __shfl_xor(s, 8, 32);  // each 16-lane half now holds its row's full sum
    int rr = row0 + r + ((lane >> 4) << 3);
    if ((lane & 15) == 0 && rr < n) out_rain[rr] = rain0[rr] + s;
  }
}

extern "C" void kernel_launch(void* const* d_in, const int* in_sizes, int n_in,
                              void* d_out, int out_size, void* d_ws, size_t ws_size,
                              hipStream_t stream) {
  const float* h_feat  = (const float*)d_in[0];
  const float* e_feat  = (const float*)d_in[1];
  const float* rain0   = (const float*)d_in[2];
  const float* w_xx    = (const float*)d_in[3];
  const float* w_yy    = (const float*)d_in[4];
  const float* W_trans = (const float*)d_in[5];
  const float* W_rin   = (const float*)d_in[6];
  const float* W_rout  = (const float*)d_in[7];
  const int*   src_xx  = (const int*)d_in[8];
  const int*   dst_xx  = (const int*)d_in[9];
  const int*   src_yy  = (const int*)d_in[10];
  const int*   dst_yy  = (const int*)d_in[11];

  const int n = in_sizes[2];   // rain0 length = N
  const int E = in_sizes[3];   // w_xx length = E

  float* out_rain = (float*)d_out;
  float* out_h    = out_rain + n;

  // workspace layout (floats): emh[32n] | h_sum[32n] | gx[16n] | gy[16n] | g2x[16n] | g2y[16n]
  float* emh   = (float*)d_ws;
  float* h_sum = emh   + (size_t)32 * n;
  float* gx    = h_sum + (size_t)32 * n;
  float* gy    = gx    + (size_t)16 * n;
  float* g2x   = gy    + (size_t)16 * n;
  float* g2y   = g2x   + (size_t)16 * n;

  // 1) zero the 96n accumulator floats (h_sum .. g2y) every call
  {
    int zn = 96 * n;
    zero_kernel<<<(zn + 255) / 256, 256, 0, stream>>>(h_sum, zn);
  }

  // 2) emh = relu([h|e] @ W_trans) via f32 WMMA (8 waves/block)
  int ntiles = (n + 15) / 16;
  {
    int blocks = (ntiles + 7) / 8;
    emh_kernel<<<blocks, 256, 0, stream>>>(h_feat, e_feat, W_trans, emh, n);
  }

  // 3) edge pass 1: h_sum + grad_x/grad_y scatter-adds (both edge sets)
  {
    long long threads = 2LL * E * 32;
    long long blocks = (threads + 255) / 256;
    edge_pass1<<<(unsigned)blocks, 256, 0, stream>>>(
        emh, e_feat, src_xx, dst_xx, w_xx, src_yy, dst_yy, w_yy,
        h_sum, gx, gy, E);
  }

  // 4) edge pass 2: grad2_x/grad2_y (serialized after pass 1 by stream order)
  {
    long long threads = 2LL * E * 16;
    long long blocks = (threads + 255) / 256;
    edge_pass2<<<(unsigned)blocks, 256, 0, stream>>>(
        gx, gy, src_xx, dst_xx, w_xx, src_yy, dst_yy, w_yy, g2x, g2y, E);
  }

  // 5) final MLP head via f32 WMMA + LDS-staged weights (8 waves/block)
  {
    int blocks = (ntiles + 7) / 8;
    rain_kernel<<<blocks, 256, 0, stream>>>(
        h_sum, gx, gy, g2x, g2y, W_rin, W_rout, rain0, out_rain, out_h, n);
  }
}